// predictor_neural_net_62380105007155
// MI455X (gfx1250) — compile-verified
//
#include <hip/hip_runtime.h>
#include <hip/hip_bf16.h>
#include <math.h>

typedef float f32x2 __attribute__((ext_vector_type(2)));
typedef float f32x8 __attribute__((ext_vector_type(8)));

#define ACT_NONE 0
#define ACT_TANH 1

// ---------------------------------------------------------------------------
// Degree accumulation: outdeg[src[e]] += 1, indeg[dst[e]] += 1
// ---------------------------------------------------------------------------
__global__ void degree_kernel(const int* __restrict__ src, const int* __restrict__ dst,
                              float* __restrict__ outdeg, float* __restrict__ indeg, int E) {
    int i = blockIdx.x * blockDim.x + threadIdx.x;
    if (i < E) {
        atomicAdd(&outdeg[src[i]], 1.0f);
        atomicAdd(&indeg[dst[i]], 1.0f);
    }
}

// deg -> rsqrt(max(deg, 1))   (in place, both arrays)
__global__ void invsqrt_kernel(float* __restrict__ outdeg, float* __restrict__ indeg, int N) {
    int i = blockIdx.x * blockDim.x + threadIdx.x;
    if (i < N) {
        outdeg[i] = rsqrtf(fmaxf(outdeg[i], 1.0f));
        indeg[i]  = rsqrtf(fmaxf(indeg[i], 1.0f));
    }
}

// ---------------------------------------------------------------------------
// fp32 WMMA GEMM:  C[nrows x Dout] = (rowscale * A)[nrows x K] @ W[K x Dout]
// Optional bias + tanh epilogue. One wave per 16x16 output tile; 8 waves per
// block share a K x 16 column panel of W staged in LDS.
// Fragment layouts per CDNA5 ISA 7.12.2 (V_WMMA_F32_16X16X4_F32):
//   A 16x4 : lanes 0-15 -> M=lane, v0=K0,v1=K1 ; lanes 16-31 -> v0=K2,v1=K3
//   B 4x16 : lanes 0-15 -> N=lane, v0=K0,v1=K1 ; lanes 16-31 -> v0=K2,v1=K3
//   D      : VGPR v -> M=v (lanes 0-15) / M=v+8 (lanes 16-31), N=lane&15
// ---------------------------------------------------------------------------
__global__ __launch_bounds__(256) void gemm_wmma_f32(
    const float* __restrict__ A, const float* __restrict__ W,
    const float* __restrict__ rowscale, const float* __restrict__ bias,
    float* __restrict__ C, int nrows, int K, int Dout, int act)
{
    __shared__ float panel[256 * 16];            // K (<=256) x 16 column panel of W

    const int lane = threadIdx.x & 31;
    const int wave = threadIdx.x >> 5;
    const int half = lane >> 4;                  // 0: lanes 0-15, 1: lanes 16-31
    const int l16  = lane & 15;

    const int rowTiles = nrows >> 4;             // nrows is a multiple of 16
    const int rt       = blockIdx.x * 8 + wave;  // this wave's row tile
    const bool active  = (rt < rowTiles);        // wave-uniform -> EXEC all-ones inside

    const float* Arow = nullptr;
    float s = 1.0f;
    if (active) {
        const int arow = rt * 16 + l16;          // row this lane supplies for A frags
        Arow = A + (size_t)arow * K;
        if (rowscale) s = rowscale[arow];
    }

    const int colTiles = Dout >> 4;
    for (int ct = 0; ct < colTiles; ++ct) {
        __syncthreads();
        // cooperative load of W[:, ct*16 .. ct*16+15] into LDS
        for (int idx = threadIdx.x; idx < K * 16; idx += 256) {
            const int k = idx >> 4;
            const int c = idx & 15;
            panel[idx] = W[(size_t)k * Dout + ct * 16 + c];
        }
        __syncthreads();

        if (active) {
            f32x8 acc = {0.f, 0.f, 0.f, 0.f, 0.f, 0.f, 0.f, 0.f};
            for (int kb = 0; kb < K; kb += 4) {
                const int k0 = kb + half * 2;
                f32x2 a, b;
                a.x = Arow[k0]     * s;
                a.y = Arow[k0 + 1] * s;
                b.x = panel[k0 * 16 + l16];
                b.y = panel[(k0 + 1) * 16 + l16];
                acc = __builtin_amdgcn_wmma_f32_16x16x4_f32(
                        /*neg_a=*/false, a, /*neg_b=*/false, b,
                        /*c_mod=*/(short)0, acc, /*reuse_a=*/false, /*reuse_b=*/false);
            }
            const float bv = bias ? bias[ct * 16 + l16] : 0.0f;
            #pragma unroll
            for (int v = 0; v < 8; ++v) {
                const int mrow = rt * 16 + v + half * 8;
                float val = acc[v] + bv;
                if (act == ACT_TANH) val = tanhf(val);
                C[(size_t)mrow * Dout + ct * 16 + l16] = val;
            }
        }
    }
}

// ---------------------------------------------------------------------------
// SpMM aggregation: agg[dst[e], :] += T[src[e], :]  (one wave per edge)
// ---------------------------------------------------------------------------
__global__ __launch_bounds__(256) void spmm_agg(
    const float* __restrict__ T, const int* __restrict__ src,
    const int* __restrict__ dst, float* __restrict__ agg, int E, int D)
{
    const long long gtid = (long long)blockIdx.x * blockDim.x + threadIdx.x;
    const int e    = (int)(gtid >> 5);
    const int lane = threadIdx.x & 31;
    if (e >= E) return;
    const int s = src[e];
    const int d = dst[e];
    const float* ts = T + (size_t)s * D;
    float* ad       = agg + (size_t)d * D;
    for (int f = lane; f < D; f += 32)
        atomicAdd(&ad[f], ts[f]);
}

// ---------------------------------------------------------------------------
// out = tanh(agg * indeg_inv[row] + bias[col])   elementwise over nrows x D
// ---------------------------------------------------------------------------
__global__ void finalize_tanh(const float* __restrict__ agg,
                              const float* __restrict__ indeg_inv,
                              const float* __restrict__ bias,
                              float* __restrict__ O, int nrows, int D)
{
    const long long i = (long long)blockIdx.x * blockDim.x + threadIdx.x;
    const long long total = (long long)nrows * D;
    if (i >= total) return;
    const int row = (int)(i / D);
    const int col = (int)(i % D);
    O[i] = tanhf(agg[i] * indeg_inv[row] + bias[col]);
}

// ---------------------------------------------------------------------------
// Row softmax over D=128 with optional rowscale + bias applied first.
// One wave per row, 4 elements per lane, shfl_xor reductions (wave32).
// ---------------------------------------------------------------------------
__global__ __launch_bounds__(256) void softmax128(
    const float* __restrict__ X, const float* __restrict__ rowscale,
    const float* __restrict__ bias, float* __restrict__ O, int nrows)
{
    const int lane = threadIdx.x & 31;
    const int wave = threadIdx.x >> 5;
    const int row  = blockIdx.x * 8 + wave;
    if (row >= nrows) return;

    const float s = rowscale ? rowscale[row] : 1.0f;
    const float* xr = X + (size_t)row * 128;

    float v[4];
    float m = -3.402823466e38f;
    #pragma unroll
    for (int j = 0; j < 4; ++j) {
        const int col = lane * 4 + j;
        const float b = bias ? bias[col] : 0.0f;
        v[j] = xr[col] * s + b;
        m = fmaxf(m, v[j]);
    }
    #pragma unroll
    for (int off = 16; off >= 1; off >>= 1)
        m = fmaxf(m, __shfl_xor(m, off, 32));

    float sum = 0.0f;
    #pragma unroll
    for (int j = 0; j < 4; ++j) { v[j] = expf(v[j] - m); sum += v[j]; }
    #pragma unroll
    for (int off = 16; off >= 1; off >>= 1)
        sum += __shfl_xor(sum, off, 32);

    const float inv = 1.0f / sum;
    float* orow = O + (size_t)row * 128;
    #pragma unroll
    for (int j = 0; j < 4; ++j)
        orow[lane * 4 + j] = v[j] * inv;
}

// ---------------------------------------------------------------------------
// Host-side orchestration
// ---------------------------------------------------------------------------
extern "C" void kernel_launch(void* const* d_in, const int* in_sizes, int n_in,
                              void* d_out, int out_size, void* d_ws, size_t ws_size,
                              hipStream_t stream)
{
    (void)n_in; (void)out_size; (void)ws_size;

    const int D_IN = 128, H = 256, D_OUT = 128;
    const int N = in_sizes[0] / D_IN;   // 100000
    const int E = in_sizes[1];          // 1600000

    const float* in_feat = (const float*)d_in[0];
    const int*   src     = (const int*)  d_in[1];
    const int*   dst     = (const int*)  d_in[2];
    const float* Wc0 = (const float*)d_in[3];  const float* bc0 = (const float*)d_in[4];
    const float* Wc1 = (const float*)d_in[5];  const float* bc1 = (const float*)d_in[6];
    const float* Wc2 = (const float*)d_in[7];  const float* bc2 = (const float*)d_in[8];
    const float* Wr0 = (const float*)d_in[9];  const float* br0 = (const float*)d_in[10];
    const float* Wr1 = (const float*)d_in[11]; const float* br1 = (const float*)d_in[12];
    const float* Wr2 = (const float*)d_in[13]; const float* br2 = (const float*)d_in[14];

    float* out = (float*)d_out;
    float* h_out = out;                          // N x 128
    float* r_out = out + (size_t)N * D_OUT;      // N x 128

    // workspace layout
    float* degO = (float*)d_ws;                  // N   (becomes outdeg^-1/2)
    float* degI = degO + N;                      // N   (becomes indeg^-1/2)
    float* bufA = degI + N;                      // N x 256
    float* bufB = bufA + (size_t)N * H;          // N x 256

    const int tpb = 256;
    const int gemmBlocks  = ((N / 16) + 7) / 8;
    const int spmmBlocks  = (int)(((long long)E * 32 + tpb - 1) / tpb);
    const int elemBlocksH = (int)(((long long)N * H + tpb - 1) / tpb);
    const int rowBlocks   = (N + 7) / 8;

    // degrees
    hipMemsetAsync(degO, 0, (size_t)2 * N * sizeof(float), stream);
    degree_kernel<<<(E + tpb - 1) / tpb, tpb, 0, stream>>>(src, dst, degO, degI, E);
    invsqrt_kernel<<<(N + tpb - 1) / tpb, tpb, 0, stream>>>(degO, degI, N);

    // ---- GraphConv 0: 128 -> 256, tanh -------------------------------------
    gemm_wmma_f32<<<gemmBlocks, tpb, 0, stream>>>(in_feat, Wc0, degO, nullptr,
                                                  bufA, N, D_IN, H, ACT_NONE);
    hipMemsetAsync(bufB, 0, (size_t)N * H * sizeof(float), stream);
    spmm_agg<<<spmmBlocks, tpb, 0, stream>>>(bufA, src, dst, bufB, E, H);
    finalize_tanh<<<elemBlocksH, tpb, 0, stream>>>(bufB, degI, bc0, bufA, N, H);

    // ---- GraphConv 1: 256 -> 256, tanh -------------------------------------
    gemm_wmma_f32<<<gemmBlocks, tpb, 0, stream>>>(bufA, Wc1, degO, nullptr,
                                                  bufB, N, H, H, ACT_NONE);
    hipMemsetAsync(bufA, 0, (size_t)N * H * sizeof(float), stream);
    spmm_agg<<<spmmBlocks, tpb, 0, stream>>>(bufB, src, dst, bufA, E, H);
    finalize_tanh<<<elemBlocksH, tpb, 0, stream>>>(bufA, degI, bc1, bufB, N, H);

    // ---- GraphConv 2: 256 -> 128, softmax ----------------------------------
    gemm_wmma_f32<<<gemmBlocks, tpb, 0, stream>>>(bufB, Wc2, degO, nullptr,
                                                  bufA, N, H, D_OUT, ACT_NONE);
    hipMemsetAsync(bufB, 0, (size_t)N * D_OUT * sizeof(float), stream);
    spmm_agg<<<spmmBlocks, tpb, 0, stream>>>(bufA, src, dst, bufB, E, D_OUT);
    softmax128<<<rowBlocks, tpb, 0, stream>>>(bufB, degI, bc2, h_out, N);

    // ---- Readout MLP -------------------------------------------------------
    gemm_wmma_f32<<<gemmBlocks, tpb, 0, stream>>>(h_out, Wr0, nullptr, br0,
                                                  bufA, N, D_OUT, H, ACT_TANH);
    gemm_wmma_f32<<<gemmBlocks, tpb, 0, stream>>>(bufA, Wr1, nullptr, br1,
                                                  bufB, N, H, H, ACT_TANH);
    gemm_wmma_f32<<<gemmBlocks, tpb, 0, stream>>>(bufB, Wr2, nullptr, br2,
                                                  bufA, N, H, D_IN, ACT_NONE);
    softmax128<<<rowBlocks, tpb, 0, stream>>>(bufA, nullptr, nullptr, r_out, N);
}